// SparseMoE_29738353557802
// MI455X (gfx1250) — compile-verified
//
#include <hip/hip_runtime.h>
#include <math.h>

#define DDIM 1024          // embed dim
#define HDIM 4096          // hidden dim (4*D)
#define NEXP 8
#define TTOK 8192          // B*S tokens
#define KH   2048          // hidden chunk held in LDS per pass
#define HSTR 2052          // padded LDS row stride (floats); 2052 % 64 == 4 -> conflict-free b64 reads

typedef float v2f __attribute__((ext_vector_type(2)));
typedef float v8f __attribute__((ext_vector_type(8)));

static __device__ __forceinline__ v8f v8f_zero() {
  v8f z = {0.f, 0.f, 0.f, 0.f, 0.f, 0.f, 0.f, 0.f};
  return z;
}

// D = A(16x4 f32) * B(4x16 f32) + C(16x16 f32)  — exact fp32 matrix op on CDNA5
static __device__ __forceinline__ v8f wmma_f32(v2f a, v2f b, v8f c) {
  return __builtin_amdgcn_wmma_f32_16x16x4_f32(false, a, false, b, (short)0, c, false, false);
}

// ---------------------------------------------------------------------------
// Kernel 0: zero the output accumulator and per-expert counters
// ---------------------------------------------------------------------------
__global__ void moe_init(float* __restrict__ out, int* __restrict__ cnt, int total) {
  int i = blockIdx.x * blockDim.x + threadIdx.x;
  int stride = gridDim.x * blockDim.x;
  for (int j = i; j < total; j += stride) out[j] = 0.f;
  if (i < NEXP) cnt[i] = 0;
}

// ---------------------------------------------------------------------------
// Kernel 1: router. One wave per token. Computes noisy top-2 gating, the
// 2-way softmax probs, and compacts (token, prob) into per-expert lists.
// ---------------------------------------------------------------------------
__global__ __launch_bounds__(256) void moe_router(
    const float* __restrict__ x, const float* __restrict__ noise_u,
    const float* __restrict__ Wr, const float* __restrict__ Wn,
    const float* __restrict__ bias,
    int* __restrict__ cnt, int* __restrict__ toks, float* __restrict__ plist) {
  const int lane = threadIdx.x & 31;
  const int wave = threadIdx.x >> 5;
  const int t = blockIdx.x * 8 + wave;

  // token's x row, strided across the wave (coalesced)
  float xr[32];
#pragma unroll
  for (int i = 0; i < 32; ++i) xr[i] = x[(size_t)t * DDIM + i * 32 + lane];

  float lg[NEXP], nz[NEXP];
#pragma unroll
  for (int e = 0; e < NEXP; ++e) {
    const float* wr = Wr + (size_t)e * DDIM;
    const float* wn = Wn + (size_t)e * DDIM;
    float sr = 0.f, sn = 0.f;
#pragma unroll
    for (int i = 0; i < 32; ++i) {
      float xi = xr[i];
      sr = fmaf(xi, wr[i * 32 + lane], sr);
      sn = fmaf(xi, wn[i * 32 + lane], sn);
    }
#pragma unroll
    for (int off = 16; off >= 1; off >>= 1) {
      sr += __shfl_xor(sr, off, 32);
      sn += __shfl_xor(sn, off, 32);
    }
    lg[e] = sr;  // full sum in all lanes after butterfly
    nz[e] = sn;
  }

  if (lane == 0) {
    float v1 = -INFINITY, v2 = -INFINITY;
    int i1 = 0, i2 = 0;
#pragma unroll
    for (int e = 0; e < NEXP; ++e) {
      // numerically stable softplus
      float sp = fmaxf(nz[e], 0.f) + log1pf(expf(-fabsf(nz[e])));
      float v = noise_u[(size_t)t * NEXP + e] * sp + lg[e] + bias[e];
      if (v > v1) { v2 = v1; i2 = i1; v1 = v; i1 = e; }
      else if (v > v2) { v2 = v; i2 = e; }
    }
    // softmax over the two selected logits (exact 0 elsewhere)
    float e2 = expf(v2 - v1);
    float inv = 1.f / (1.f + e2);
    int p1pos = atomicAdd(&cnt[i1], 1);
    toks[i1 * TTOK + p1pos] = t;
    plist[i1 * TTOK + p1pos] = inv;
    int p2pos = atomicAdd(&cnt[i2], 1);
    toks[i2 * TTOK + p2pos] = t;
    plist[i2 * TTOK + p2pos] = e2 * inv;
  }
}

// ---------------------------------------------------------------------------
// Kernel 2: per-expert fused MLP. One workgroup (8 waves) handles a 16-token
// tile of one expert. Hidden dim processed in KH-wide chunks staged in LDS:
//   stage 1: h = gelu(x_tile[16xD] @ W1[:,chunk] + b1)  -> LDS (padded)
//   stage 2: y[16x1024] += h @ W2[chunk,:]              -> v8f accumulators
// Epilogue: out[token] += p * (y + b2) via float atomics (2 commutative adds
// per element onto a zeroed buffer -> deterministic).
// ---------------------------------------------------------------------------
__global__ __launch_bounds__(256) void moe_expert(
    const float* __restrict__ x,
    const float* __restrict__ W1, const float* __restrict__ b1,
    const float* __restrict__ W2, const float* __restrict__ b2,
    const int* __restrict__ cnt, const int* __restrict__ toks,
    const float* __restrict__ plist, float* __restrict__ out) {
  extern __shared__ float hbuf[];  // 16 * HSTR floats (~128.25 KB)

  const int e = blockIdx.y;
  const int n = cnt[e];
  const int base = blockIdx.x * 16;
  if (base >= n) return;  // block-uniform exit (EXEC stays all-ones for WMMA)
  const int nvalid = min(16, n - base);

  const int lane = threadIdx.x & 31;
  const int wave = threadIdx.x >> 5;
  const int mrow = lane & 15;
  const int khalf = (lane >> 4) << 1;  // K sub-offset: 0 (lanes 0-15) or 2 (lanes 16-31)

  // Per-lane A-side token row (pad rows clamp to a valid token; masked at store)
  int ridx = base + mrow;
  if (ridx >= n) ridx = base;
  const float* xrow = x + (size_t)toks[e * TTOK + ridx] * DDIM + khalf;

  const float* W1e = W1 + (size_t)e * DDIM * HDIM;
  const float* W2e = W2 + (size_t)e * HDIM * DDIM;
  const float* b1e = b1 + (size_t)e * HDIM;
  const float* b2e = b2 + (size_t)e * DDIM;

  v8f acc2[8];  // wave owns output cols [wave*128, wave*128+128)
#pragma unroll
  for (int i = 0; i < 8; ++i) acc2[i] = v8f_zero();

  for (int c = 0; c < HDIM / KH; ++c) {
    const int cbase = c * KH;

    // ---- stage 1: wave computes 16 n-tiles (256 hidden cols) of h ----
#pragma unroll 1
    for (int nt = 0; nt < 16; ++nt) {
      const int lcol = wave * 256 + nt * 16 + mrow;  // col within chunk
      const float* w1p = W1e + (size_t)khalf * HDIM + (cbase + lcol);
      v8f acc = v8f_zero();
#pragma unroll 4
      for (int k = 0; k < DDIM; k += 4) {
        v2f a = *(const v2f*)(xrow + k);                 // A: rows of x tile
        v2f b;                                           // B: W1 col, rows khalf..+1
        b.x = w1p[0];
        b.y = w1p[HDIM];
        acc = wmma_f32(a, b, acc);
        w1p += 4 * HDIM;
      }
      const float bb = b1e[cbase + lcol];
      const int rb = (lane < 16) ? 0 : 8;  // C/D layout: lanes 16-31 hold rows M+8
#pragma unroll
      for (int r = 0; r < 8; ++r) {
        float v = acc[r] + bb;
        v = 0.5f * v * (1.f + erff(v * 0.70710678118654752f));  // exact-erf GELU
        hbuf[(size_t)(r + rb) * HSTR + lcol] = v;
      }
    }
    __syncthreads();

    // ---- stage 2: y += h(chunk) @ W2(chunk rows) ----
    const float* w2p = W2e + (size_t)(cbase + khalf) * DDIM + wave * 128 + mrow;
    const float* hp = hbuf + (size_t)mrow * HSTR + khalf;
#pragma unroll 2
    for (int kk = 0; kk < KH; kk += 4) {
      v2f a = *(const v2f*)(hp + kk);  // conflict-free: stride 2052 spreads 64 banks
#pragma unroll
      for (int nt2 = 0; nt2 < 8; ++nt2) {
        v2f b;
        b.x = w2p[nt2 * 16];
        b.y = w2p[nt2 * 16 + DDIM];
        acc2[nt2] = wmma_f32(a, b, acc2[nt2]);
      }
      w2p += 4 * DDIM;
    }
    __syncthreads();  // WAR: next chunk overwrites hbuf
  }

  // ---- epilogue: out[token] += p * (y + b2) ----
  const int rb = (lane < 16) ? 0 : 8;
  float pr[8];
  size_t tko[8];
#pragma unroll
  for (int r = 0; r < 8; ++r) {
    int row = r + rb;
    int idx = e * TTOK + base + min(row, nvalid - 1);
    pr[r] = plist[idx];
    tko[r] = (size_t)toks[idx] * DDIM;
  }
#pragma unroll
  for (int nt2 = 0; nt2 < 8; ++nt2) {
    const int col = wave * 128 + nt2 * 16 + mrow;
    const float b2v = b2e[col];
#pragma unroll
    for (int r = 0; r < 8; ++r) {
      if (r + rb < nvalid) {  // per-lane mask on stores only (after all WMMA)
        unsafeAtomicAdd(&out[tko[r] + col], (acc2[nt2][r] + b2v) * pr[r]);
      }
    }
  }
}

// ---------------------------------------------------------------------------
extern "C" void kernel_launch(void* const* d_in, const int* in_sizes, int n_in,
                              void* d_out, int out_size, void* d_ws, size_t ws_size,
                              hipStream_t stream) {
  const float* x  = (const float*)d_in[0];
  const float* nu = (const float*)d_in[1];
  const float* Wr = (const float*)d_in[2];
  const float* Wn = (const float*)d_in[3];
  const float* bs = (const float*)d_in[4];
  const float* W1 = (const float*)d_in[5];
  const float* b1 = (const float*)d_in[6];
  const float* W2 = (const float*)d_in[7];
  const float* b2 = (const float*)d_in[8];
  float* out = (float*)d_out;

  // workspace layout: [cnt: 8 ints][pad to 256B][toks: E*T ints][plist: E*T floats]
  int* cnt = (int*)d_ws;
  int* toks = (int*)((char*)d_ws + 256);
  float* pls = (float*)((char*)d_ws + 256 + (size_t)NEXP * TTOK * sizeof(int));

  moe_init<<<2048, 256, 0, stream>>>(out, cnt, TTOK * DDIM);
  moe_router<<<TTOK / 8, 256, 0, stream>>>(x, nu, Wr, Wn, bs, cnt, toks, pls);
  dim3 grid(TTOK / 16, NEXP);
  moe_expert<<<grid, 256, (size_t)16 * HSTR * sizeof(float), stream>>>(
      x, W1, b1, W2, b2, cnt, toks, pls, out);
}